// GUIDE_Base_841813590022
// MI455X (gfx1250) — compile-verified
//
#include <hip/hip_runtime.h>
#include <hip/hip_bf16.h>
#include <math.h>

#define A_DIM 128
#define S_DIM 64

typedef __attribute__((ext_vector_type(2))) float v2f;
typedef __attribute__((ext_vector_type(8))) float v8f;
typedef long long i64;

__device__ __forceinline__ float gelu_exact(float x) {
    return 0.5f * x * (1.0f + erff(x * 0.7071067811865476f));
}

__device__ __forceinline__ void atomAddF(float* p, float v) {
    __hip_atomic_fetch_add(p, v, __ATOMIC_RELAXED, __HIP_MEMORY_SCOPE_AGENT);
}

// ---------------- small utility kernels ----------------
__global__ void k_fill(float* p, float v, int n) {
    int i = blockIdx.x * blockDim.x + threadIdx.x;
    if (i < n) p[i] = v;
}

__global__ void k_deg_edges(const i64* __restrict__ col, float* deg, int E) {
    int e = blockIdx.x * blockDim.x + threadIdx.x;
    if (e < E) atomAddF(&deg[(int)col[e]], 1.0f);
}

__global__ void k_dinv(float* deg, int n) {
    int i = blockIdx.x * blockDim.x + threadIdx.x;
    if (i < n) { float d = deg[i]; deg[i] = d > 0.f ? rsqrtf(d) : 0.f; }
}

__global__ void k_norm(const i64* __restrict__ row, const i64* __restrict__ col,
                       const float* __restrict__ dinv, float* norm, int E) {
    int e = blockIdx.x * blockDim.x + threadIdx.x;
    if (e < E) norm[e] = dinv[(int)row[e]] * dinv[(int)col[e]];
}

// ---------------- dense GEMM via fp32 WMMA ----------------
// C[N x K] = A[N x K] @ W[K x K] (+ bias per column).
// One wave computes one 16x16 tile of C; block = (K/16) waves covering a
// 16-row stripe. V_WMMA_F32_16X16X4_F32: A 16x4, B 4x16, C 16x16.
template<int K, bool BIAS>
__global__ void k_gemm_wmma(const float* __restrict__ A, const float* __restrict__ W,
                            const float* __restrict__ bias, float* __restrict__ C,
                            int Nrows) {
    const int lane  = threadIdx.x & 31;
    const int wave  = threadIdx.x >> 5;          // column tile index
    const int m0    = blockIdx.x * 16;           // row tile base
    const int n0    = wave * 16;
    const int l15   = lane & 15;
    const int khalf = (lane >> 4) << 1;          // 0 (lanes 0-15) or 2 (16-31)
    int mrow = m0 + l15;
    if (mrow >= Nrows) mrow = Nrows - 1;         // clamp (keeps EXEC all-1s)
    const int ncol = n0 + l15;

    v8f c = {};
    for (int k = 0; k < K; k += 4) {
        v2f a, b;
        const float* ap = A + (size_t)mrow * K + (k + khalf);
        a.x = ap[0];                              // A[m][k+khalf]
        a.y = ap[1];                              // A[m][k+khalf+1]
        const float* bp = W + (size_t)(k + khalf) * K + ncol;
        b.x = bp[0];                              // W[k+khalf][n]
        b.y = bp[K];                              // W[k+khalf+1][n]
        c = __builtin_amdgcn_wmma_f32_16x16x4_f32(false, a, false, b,
                                                  (short)0, c, false, false);
    }
    const int rbase = m0 + ((lane >> 4) << 3);    // rows v / v+8 per lane half
#pragma unroll
    for (int v = 0; v < 8; ++v) {
        int r = rbase + v;
        if (r < Nrows) {
            float out = c[v];
            if (BIAS) out += bias[ncol];
            C[(size_t)r * K + ncol] = out;
        }
    }
}

// ---------------- GCN aggregation ----------------
__global__ void k_gcn_selfinit(const float* __restrict__ h2, const float* __restrict__ dinv,
                               float* agg, int n) {
    int i = blockIdx.x * blockDim.x + threadIdx.x;
    if (i < n * A_DIM) {
        float d = dinv[i >> 7];
        agg[i] = d * d * h2[i];
    }
}

__global__ void k_gcn_scatter(const i64* __restrict__ row, const i64* __restrict__ col,
                              const float* __restrict__ norm, const float* __restrict__ h2,
                              float* agg, int E) {
    int t = blockIdx.x * blockDim.x + threadIdx.x;
    int e = t >> 5, lane = t & 31;
    if (e >= E) return;
    int r = (int)row[e], c = (int)col[e];
    float nw = norm[e];
    const float4 v = *(const float4*)(h2 + (size_t)r * A_DIM + lane * 4);
    float* dst = agg + (size_t)c * A_DIM + lane * 4;
    atomAddF(dst + 0, nw * v.x);
    atomAddF(dst + 1, nw * v.y);
    atomAddF(dst + 2, nw * v.z);
    atomAddF(dst + 3, nw * v.w);
}

template<bool ACT>
__global__ void k_bias_act128(float* h, const float* __restrict__ bias, int n) {
    int i = blockIdx.x * blockDim.x + threadIdx.x;
    if (i < n * A_DIM) {
        float v = h[i] + bias[i & (A_DIM - 1)];
        h[i] = ACT ? gelu_exact(v) : v;
    }
}

// ---------------- GNA attention ----------------
__global__ void k_alpha(const i64* __restrict__ row, const i64* __restrict__ col,
                        const float* __restrict__ t, const float* __restrict__ avec,
                        float* alpha, int E) {
    int tid = blockIdx.x * blockDim.x + threadIdx.x;
    int e = tid >> 5, lane = tid & 31;
    if (e >= E) return;
    int r = (int)row[e], c = (int)col[e];
    int d = lane * 2;
    float2 tc = *(const float2*)(t + (size_t)c * S_DIM + d);
    float2 tr = *(const float2*)(t + (size_t)r * S_DIM + d);
    float2 av = *(const float2*)(avec + d);
    float p = (tc.x - tr.x) * av.x + (tc.y - tr.y) * av.y;
#pragma unroll
    for (int off = 16; off > 0; off >>= 1) p += __shfl_xor(p, off, 32);
    if (lane == 0) alpha[e] = p;
}

// amax init = 0 (self-loop logit); positive floats order-match their int bits,
// and nothing below 0 ever needs to win, so int atomicMax is exact.
__global__ void k_amax_edges(const i64* __restrict__ col, const float* __restrict__ alpha,
                             float* amax, int E) {
    int e = blockIdx.x * blockDim.x + threadIdx.x;
    if (e >= E) return;
    float a = alpha[e];
    if (a > 0.0f) atomicMax((int*)&amax[(int)col[e]], __float_as_int(a));
}

__global__ void k_denom_init(const float* __restrict__ amax, float* denom, int n) {
    int i = blockIdx.x * blockDim.x + threadIdx.x;
    if (i < n) denom[i] = expf(-amax[i]);   // self-loop term exp(0 - amax)
}

__global__ void k_exp_denom(const i64* __restrict__ col, float* alpha,
                            const float* __restrict__ amax, float* denom, int E) {
    int e = blockIdx.x * blockDim.x + threadIdx.x;
    if (e >= E) return;
    int c = (int)col[e];
    float ex = expf(alpha[e] - amax[c]);
    alpha[e] = ex;
    atomAddF(&denom[c], ex);
}

__global__ void k_msg_selfinit(const float* __restrict__ t, const float* __restrict__ amax,
                               const float* __restrict__ denom, float* base, int n) {
    int i = blockIdx.x * blockDim.x + threadIdx.x;
    if (i < n * S_DIM) {
        int node = i >> 6;
        float w = expf(-amax[node]) / (denom[node] + 1e-16f);
        base[i] += w * t[i];
    }
}

__global__ void k_msg_scatter(const i64* __restrict__ row, const i64* __restrict__ col,
                              const float* __restrict__ alpha, const float* __restrict__ denom,
                              const float* __restrict__ t, float* base, int E) {
    int tid = blockIdx.x * blockDim.x + threadIdx.x;
    int e = tid >> 5, lane = tid & 31;
    if (e >= E) return;
    int r = (int)row[e], c = (int)col[e];
    float w = alpha[e] / (denom[c] + 1e-16f);
    float2 tv = *(const float2*)(t + (size_t)r * S_DIM + lane * 2);
    float* dst = base + (size_t)c * S_DIM + lane * 2;
    atomAddF(dst + 0, w * tv.x);
    atomAddF(dst + 1, w * tv.y);
}

__global__ void k_gelu(float* p, int n) {
    int i = blockIdx.x * blockDim.x + threadIdx.x;
    if (i < n) p[i] = gelu_exact(p[i]);
}

static inline int cdiv(long long a, long long b) { return (int)((a + b - 1) / b); }

extern "C" void kernel_launch(void* const* d_in, const int* in_sizes, int n_in,
                              void* d_out, int out_size, void* d_ws, size_t ws_size,
                              hipStream_t stream) {
    const float* x     = (const float*)d_in[0];
    const float* s     = (const float*)d_in[1];
    const i64*   ei    = (const i64*)d_in[2];     // int64 per reference
    const float* gcn_W = (const float*)d_in[3];
    const float* gcn_b = (const float*)d_in[4];
    const float* w1W   = (const float*)d_in[5];
    const float* w1b   = (const float*)d_in[6];
    const float* w2W   = (const float*)d_in[7];
    const float* w2b   = (const float*)d_in[8];
    const float* ga    = (const float*)d_in[9];

    const int N = in_sizes[0] / A_DIM;
    const int E = in_sizes[2] / 2;
    const i64* erow = ei;         // src
    const i64* ecol = ei + E;     // dst

    float* ws = (float*)d_ws;
    size_t off = 0;
    float* dinv  = ws + off; off += (size_t)N;
    float* h_a   = ws + off; off += (size_t)N * A_DIM;
    float* h_b   = ws + off; off += (size_t)N * A_DIM;
    float* h2    = ws + off; off += (size_t)N * A_DIM;
    float* g_a   = ws + off; off += (size_t)N * S_DIM;
    float* g_b   = ws + off; off += (size_t)N * S_DIM;
    float* tbuf  = ws + off; off += (size_t)N * S_DIM;
    float* amax  = ws + off; off += (size_t)N;
    float* denom = ws + off; off += (size_t)N;
    float* norm  = ws + off; off += (size_t)E;
    float* alpha = ws + off; off += (size_t)E;

    float* h_out = (float*)d_out;                          // N x 128
    float* g_out = (float*)d_out + (size_t)N * A_DIM;      // N x 64

    const int T = 256;

    // ---- degrees (self-loop folded into init=1) & symmetric norm ----
    k_fill<<<cdiv(N, T), T, 0, stream>>>(dinv, 1.0f, N);
    k_deg_edges<<<cdiv(E, T), T, 0, stream>>>(ecol, dinv, E);
    k_dinv<<<cdiv(N, T), T, 0, stream>>>(dinv, N);
    k_norm<<<cdiv(E, T), T, 0, stream>>>(erow, ecol, dinv, norm, E);

    // ---- GCN branch ----
    const float* hcur = x;
    float* hseq[4] = { h_a, h_b, h_a, h_out };
    for (int l = 0; l < 4; ++l) {
        float* hnext = hseq[l];
        k_gemm_wmma<A_DIM, false><<<cdiv(N, 16), 32 * (A_DIM / 16), 0, stream>>>(
            hcur, gcn_W + (size_t)l * A_DIM * A_DIM, nullptr, h2, N);
        k_gcn_selfinit<<<cdiv((long long)N * A_DIM, T), T, 0, stream>>>(h2, dinv, hnext, N);
        k_gcn_scatter<<<cdiv((long long)E * 32, T), T, 0, stream>>>(erow, ecol, norm, h2, hnext, E);
        if (l < 3)
            k_bias_act128<true ><<<cdiv((long long)N * A_DIM, T), T, 0, stream>>>(hnext, gcn_b + (size_t)l * A_DIM, N);
        else
            k_bias_act128<false><<<cdiv((long long)N * A_DIM, T), T, 0, stream>>>(hnext, gcn_b + (size_t)l * A_DIM, N);
        hcur = hnext;
    }

    // ---- GNA branch ----
    const float* gcur = s;
    float* gseq[4] = { g_a, g_b, g_a, g_out };
    for (int l = 0; l < 4; ++l) {
        float* gnext = gseq[l];
        // t = g @ w2 + b2 ; base = g @ w1 + b1 (into gnext)
        k_gemm_wmma<S_DIM, true><<<cdiv(N, 16), 32 * (S_DIM / 16), 0, stream>>>(
            gcur, w2W + (size_t)l * S_DIM * S_DIM, w2b + (size_t)l * S_DIM, tbuf, N);
        k_gemm_wmma<S_DIM, true><<<cdiv(N, 16), 32 * (S_DIM / 16), 0, stream>>>(
            gcur, w1W + (size_t)l * S_DIM * S_DIM, w1b + (size_t)l * S_DIM, gnext, N);
        // edge softmax over incoming edges of each dst (self-loop logit = 0)
        k_alpha<<<cdiv((long long)E * 32, T), T, 0, stream>>>(erow, ecol, tbuf,
                                                              ga + (size_t)l * S_DIM, alpha, E);
        k_fill<<<cdiv(N, T), T, 0, stream>>>(amax, 0.0f, N);
        k_amax_edges<<<cdiv(E, T), T, 0, stream>>>(ecol, alpha, amax, E);
        k_denom_init<<<cdiv(N, T), T, 0, stream>>>(amax, denom, N);
        k_exp_denom<<<cdiv(E, T), T, 0, stream>>>(ecol, alpha, amax, denom, E);
        // messages: self-loop + weighted neighbor t[row]
        k_msg_selfinit<<<cdiv((long long)N * S_DIM, T), T, 0, stream>>>(tbuf, amax, denom, gnext, N);
        k_msg_scatter<<<cdiv((long long)E * 32, T), T, 0, stream>>>(erow, ecol, alpha, denom, tbuf, gnext, E);
        k_gelu<<<cdiv((long long)N * S_DIM, T), T, 0, stream>>>(gnext, N * S_DIM);
        gcur = gnext;
    }
}